// MyLoss_19473381720640
// MI455X (gfx1250) — compile-verified
//
#include <hip/hip_runtime.h>

typedef float v2f __attribute__((ext_vector_type(2)));
typedef float v4f __attribute__((ext_vector_type(4)));
typedef float v8f __attribute__((ext_vector_type(8)));

// ---------------------------------------------------------------------------
// Kernel 1: zero the double accumulator in workspace (harness poisons ws).
// ---------------------------------------------------------------------------
__global__ void zero_acc_kernel(double* acc) {
    if (threadIdx.x == 0 && blockIdx.x == 0) acc[0] = 0.0;
}

// ---------------------------------------------------------------------------
// Kernel 2: bandwidth-bound streaming reduction.
//   partial = sum_i log(output[i]) * label[i]
// Non-temporal b128 loads (823 MB streamed once, > L2, so NT is correct).
// Intra-wave32 reduction via V_WMMA_F32_16X16X4_F32 (A = per-lane partials,
// B = ones), then one shfl_xor(16), then one f64 atomic per wave.
// ---------------------------------------------------------------------------
__global__ __launch_bounds__(256) void
logdot_reduce_kernel(const v4f* __restrict__ out4,
                     const v4f* __restrict__ lab4,
                     long n4, double* __restrict__ acc) {
    const long tid    = (long)blockIdx.x * blockDim.x + threadIdx.x;
    const long stride = (long)gridDim.x * blockDim.x;

    // Two accumulators per lane -> exactly the 2 VGPRs of the 16x4 A operand.
    float a0 = 0.0f, a1 = 0.0f;

    for (long i = tid; i < n4; i += stride) {
        v4f o = __builtin_nontemporal_load(&out4[i]);  // global_load_b128 TH=NT
        v4f l = __builtin_nontemporal_load(&lab4[i]);  // global_load_b128 TH=NT
        a0 = fmaf(__logf(o.x), l.x, a0);               // v_log_f32 + v_fma_f32
        a1 = fmaf(__logf(o.y), l.y, a1);
        a0 = fmaf(__logf(o.z), l.z, a0);
        a1 = fmaf(__logf(o.w), l.w, a1);
    }

    // ---- wave32 reduction on the matrix pipe ----
    // A (16x4 f32): lanes 0-15 supply K=0,1; lanes 16-31 supply K=2,3.
    // B = all-ones => D[m][n] = sum_k A[m][k] (row sums, replicated over n).
    v2f A = {a0, a1};
    v2f B = {1.0f, 1.0f};
    v8f C = {};
    v8f D = __builtin_amdgcn_wmma_f32_16x16x4_f32(
        /*neg_a=*/false, A, /*neg_b=*/false, B,
        /*c_mod=*/(short)0, C, /*reuse_a=*/false, /*reuse_b=*/false);

    // Lane d VGPRs hold rows {0..7} (lanes 0-15) or rows {8..15} (lanes 16-31).
    float s = D[0] + D[1] + D[2] + D[3] + D[4] + D[5] + D[6] + D[7];
    s += __shfl_xor(s, 16, 32);        // combine the two half-wave row groups

    if ((threadIdx.x & 31) == 0)
        atomicAdd(acc, (double)s);      // f64 accumulation: order-insensitive
}

// ---------------------------------------------------------------------------
// Kernel 3: out = -(float)acc
// ---------------------------------------------------------------------------
__global__ void finalize_kernel(const double* __restrict__ acc,
                                float* __restrict__ out) {
    if (threadIdx.x == 0 && blockIdx.x == 0) out[0] = -(float)acc[0];
}

extern "C" void kernel_launch(void* const* d_in, const int* in_sizes, int n_in,
                              void* d_out, int out_size, void* d_ws, size_t ws_size,
                              hipStream_t stream) {
    const v4f* out4 = (const v4f*)d_in[0];   // "output": (B,V) f32, B*V % 4 == 0
    const v4f* lab4 = (const v4f*)d_in[1];   // "label":  (B,V) f32
    float* out = (float*)d_out;
    double* acc = (double*)d_ws;

    const long n4 = (long)in_sizes[0] / 4;   // 2048*50257/4 = 25,731,584 float4s

    zero_acc_kernel<<<1, 64, 0, stream>>>(acc);

    // 2048 blocks x 256 threads = 512K lanes, ~49 b128-pairs each:
    // enough waves in flight to saturate 23.3 TB/s.
    int blocks = (int)((n4 + 255) / 256);
    if (blocks > 2048) blocks = 2048;
    if (blocks < 1) blocks = 1;
    logdot_reduce_kernel<<<blocks, 256, 0, stream>>>(out4, lab4, n4, acc);

    finalize_kernel<<<1, 64, 0, stream>>>(acc, out);
}